// Denoise1_14285061227117
// MI455X (gfx1250) — compile-verified
//
#include <hip/hip_runtime.h>
#include <hip/hip_bf16.h>

// LIF constants (from the reference):
//   v_decayed = v + DT*TAU_MEM_INV*(i - v)   ; DT*TAU_MEM_INV = 0.2
//   i_decayed = i - DT*TAU_SYN_INV*i         ; DT*TAU_SYN_INV = 0.4
//   z = (v_decayed - 1.0 > 0)
//   v' = (1-z)*v_decayed ; i' = i_decayed + input
#define KM 0.2f   // DT * TAU_MEM_INV
#define KS 0.4f   // DT * TAU_SYN_INV
#define VTH 1.0f

typedef __attribute__((ext_vector_type(4))) float v4f;

// Streaming 2-layer LIF scan. One thread owns 4 contiguous spatial elements,
// keeps all 16 state floats in VGPRs, and walks t = 0..T-1 with coalesced
// b128 non-temporal loads/stores (512MB stream touched exactly once -> NT
// avoids L2 thrash) plus global_prefetch_b8 a dozen timesteps ahead.
// T is a compile-time constant so there is no remainder loop and the body
// can be software-pipelined with 8 independent loads in flight per wave.
template <int T>
__global__ __launch_bounds__(256) void lif2_scan_kernel(
    const float* __restrict__ x,   // [T, N] flattened
    float* __restrict__ out,       // [T, N] flattened
    int N) {
  const int gid = blockIdx.x * blockDim.x + threadIdx.x;  // vec4 index
  const int base = gid * 4;
  if (base >= N) return;

  // Prefetch 12 timesteps (= 12 MB across the grid) ahead; only 1 of every
  // 8 lanes issues it so each enabled lane covers a distinct 128B line of
  // the wave's 512B-per-timestep footprint.
  constexpr int PF = 12;
  const bool do_pref = ((threadIdx.x & 7u) == 0u);

  float v1[4], i1[4], v2[4], i2[4];
#pragma unroll
  for (int k = 0; k < 4; ++k) { v1[k] = 0.f; i1[k] = 0.f; v2[k] = 0.f; i2[k] = 0.f; }

  int off = base;  // element offset; max (T-1)*N + base < 2^27, fits int
#pragma unroll 8
  for (int t = 0; t < T; ++t, off += N) {
    if (do_pref && (t + PF) < T) {
      __builtin_prefetch(x + off + PF * N, 0, 1);  // -> global_prefetch_b8
    }

    // Non-temporal 128-bit streaming load (th:NT)
    const v4f xin = __builtin_nontemporal_load((const v4f*)(x + off));

    v4f z;
#pragma unroll
    for (int k = 0; k < 4; ++k) {
      // ---- layer 1 ----
      float vd1 = __builtin_fmaf(KM, i1[k] - v1[k], v1[k]);  // v + 0.2*(i-v)
      float id1 = __builtin_fmaf(-KS, i1[k], i1[k]);          // i - 0.4*i
      const bool s1 = (vd1 - VTH) > 0.0f;                     // heaviside
      const float z1 = s1 ? 1.0f : 0.0f;
      v1[k] = s1 ? 0.0f : vd1;                                // (1-z)*v_decayed
      i1[k] = id1 + xin[k];

      // ---- layer 2 (input = z1) ----
      float vd2 = __builtin_fmaf(KM, i2[k] - v2[k], v2[k]);
      float id2 = __builtin_fmaf(-KS, i2[k], i2[k]);
      const bool s2 = (vd2 - VTH) > 0.0f;
      z[k] = s2 ? 1.0f : 0.0f;
      v2[k] = s2 ? 0.0f : vd2;
      i2[k] = id2 + z1;
    }

    // Non-temporal 128-bit streaming store (th:NT)
    __builtin_nontemporal_store(z, (v4f*)(out + off));
  }
}

extern "C" void kernel_launch(void* const* d_in, const int* in_sizes, int n_in,
                              void* d_out, int out_size, void* d_ws, size_t ws_size,
                              hipStream_t stream) {
  const float* x = (const float*)d_in[0];
  float* out = (float*)d_out;

  constexpr int T = 256;                   // reference: x is [256, 16, 128, 128]
  const int total = in_sizes[0];           // 67,108,864
  const int N = total / T;                 // 262,144 spatial elements
  const int nvec = N / 4;                  // 65,536 threads (2048 wave32s)

  dim3 block(256);
  dim3 grid((nvec + block.x - 1) / block.x);
  lif2_scan_kernel<T><<<grid, block, 0, stream>>>(x, out, N);
}